// GetStoneDistAngle3d_53635551592643
// MI455X (gfx1250) — compile-verified
//
#include <hip/hip_runtime.h>
#include <math.h>

// CDNA5 WMMA vector operand types (wave32):
//  A 16x4 f32  -> 64 vals / 32 lanes = 2 VGPRs -> float2
//  B 4x16 f32  -> 64 vals / 32 lanes = 2 VGPRs -> float2
//  C/D 16x16 f32 -> 256 vals / 32 lanes = 8 VGPRs -> float8
typedef __attribute__((ext_vector_type(2))) float v2f;
typedef __attribute__((ext_vector_type(8))) float v8f;

#define SLEN 4096         // stones per row == bitonic sort length (power of 2)
#define ROWS_PER_WG 16    // board rows handled per workgroup (== WMMA M)
#define TPB 1024          // 32 wave32s
#define RAD2DEG 57.29577951308232f

__global__ __launch_bounds__(TPB)
void GetStoneDistAngle3d_kernel(const float* __restrict__ coords,  // [N,3] flag,y,x
                                const float* __restrict__ stones,  // [S,3] val,y,x
                                float* __restrict__ out,           // [N,S,3]
                                int N, int S)
{
    extern __shared__ unsigned char smem[];
    float* s_d2 = (float*)smem;                               // 16*4096 f32 = 256 KB
    unsigned long long* s_key =
        (unsigned long long*)(smem + (size_t)ROWS_PER_WG * SLEN * sizeof(float)); // 32 KB

    const int tid   = threadIdx.x;
    const int lane  = tid & 31;
    const int wave  = tid >> 5;
    const int nwave = TPB >> 5;
    const int rowBase = blockIdx.x * ROWS_PER_WG;

    // Pull the (L2-hot) stone table toward this WGP: emits global_prefetch_b8.
    if (tid < 64) __builtin_prefetch(stones + tid * 192, 0, 0);

    // ---------------- Phase 1: dist^2 Gram matrix via V_WMMA_F32_16X16X4_F32 ----
    // dist^2[m][n] = cy*(-2sy) + cx*(-2sx) + |c|^2*1 + 1*|s|^2
    const int  m  = lane & 15;
    const bool hi = lane >= 16;

    int rowM = rowBase + m;
    float cy = 0.0f, cx = 0.0f;
    if (rowM < N) { cy = coords[rowM * 3 + 1]; cx = coords[rowM * 3 + 2]; }
    const float cc = cy * cy + cx * cx;

    // A layout: lanes 0-15 hold (K0,K1)=(cy,cx); lanes 16-31 hold (K2,K3)=(|c|^2,1)
    v2f a;
    a.x = hi ? cc   : cy;
    a.y = hi ? 1.0f : cx;

    const int ntiles = SLEN / 16;
    for (int t = wave; t < ntiles; t += nwave) {
        int sIdx = t * 16 + m;
        float sy = 0.0f, sx = 0.0f;
        if (sIdx < S) { sy = stones[sIdx * 3 + 1]; sx = stones[sIdx * 3 + 2]; }
        const float ss = sy * sy + sx * sx;

        // B layout: lanes 0-15 hold rows (K0,K1)=(-2sy,-2sx); lanes 16-31 hold (K2,K3)=(1,|s|^2)
        v2f b;
        b.x = hi ? 1.0f : (-2.0f * sy);
        b.y = hi ? ss   : (-2.0f * sx);

        v8f c = {};
        v8f d = __builtin_amdgcn_wmma_f32_16x16x4_f32(
            /*neg_a=*/false, a, /*neg_b=*/false, b,
            /*c_mod=*/(short)0, c, /*reuse_a=*/false, /*reuse_b=*/false);

        // D layout: VGPR v -> M = v + (hi?8:0), N = lane&15
        const int mOff = hi ? 8 : 0;
        #pragma unroll
        for (int v = 0; v < 8; ++v) {
            s_d2[(v + mOff) * SLEN + t * 16 + m] = d[v];
        }
    }
    __syncthreads();

    // ---------------- Phase 2: per-row stable sort by distance + emit ----------
    for (int r = 0; r < ROWS_PER_WG; ++r) {
        int gRow = rowBase + r;
        if (gRow >= N) break;

        const float flag = coords[gRow * 3 + 0];
        const float bcy  = coords[gRow * 3 + 1];
        const float bcx  = coords[gRow * 3 + 2];
        float* orow = out + (size_t)gRow * SLEN * 3;

        if (flag != 0.0f) {                 // block-uniform branch: zero row
            for (int i = tid; i < SLEN * 3; i += TPB) orow[i] = 0.0f;
            continue;
        }

        // Build stable sort keys: dist^2 bits (monotone for x>=0) | stone index.
        for (int i = tid; i < SLEN; i += TPB) {
            float d2 = fmaxf(s_d2[r * SLEN + i], 0.0f);
            s_key[i] = ((unsigned long long)__float_as_uint(d2) << 32) | (unsigned)i;
        }
        __syncthreads();

        // Bitonic sort, ascending, 4096 u64 keys in LDS.
        for (int k = 2; k <= SLEN; k <<= 1) {
            for (int j = k >> 1; j > 0; j >>= 1) {
                for (int t = tid; t < SLEN; t += TPB) {
                    int p = t ^ j;
                    if (p > t) {
                        unsigned long long x = s_key[t];
                        unsigned long long y = s_key[p];
                        bool asc = ((t & k) == 0);
                        if ((x > y) == asc) { s_key[t] = y; s_key[p] = x; }
                    }
                }
                __syncthreads();
            }
        }

        // Emit (val, dist, angle) in sorted order; coalesced 12B/thread stores.
        for (int i = tid; i < SLEN; i += TPB) {
            unsigned long long key = s_key[i];
            int   sIdx = (int)(key & 0xFFFFFFFFull);
            float d2   = __uint_as_float((unsigned)(key >> 32));
            float dist = sqrtf(d2);
            float val  = stones[sIdx * 3 + 0];
            float sy   = stones[sIdx * 3 + 1];
            float sx   = stones[sIdx * 3 + 2];
            float dy = sy - bcy, dx = sx - bcx;
            float raw = atan2f(-dy, dx) * RAD2DEG;
            float ang = raw > 0.0f ? raw : 360.0f + raw;
            orow[i * 3 + 0] = val;
            orow[i * 3 + 1] = dist;
            orow[i * 3 + 2] = ang;
        }
        __syncthreads();   // keys may be overwritten next iteration
    }
}

extern "C" void kernel_launch(void* const* d_in, const int* in_sizes, int n_in,
                              void* d_out, int out_size, void* d_ws, size_t ws_size,
                              hipStream_t stream) {
    const float* coords = (const float*)d_in[0];   // all_coord_input  [N,3]
    const float* stones = (const float*)d_in[1];   // stone_coord_input [S,3]
    float* out = (float*)d_out;                    // [N,S,3] f32

    int N = in_sizes[0] / 3;
    int S = in_sizes[1] / 3;

    int nblocks = (N + ROWS_PER_WG - 1) / ROWS_PER_WG;
    size_t shmem = (size_t)ROWS_PER_WG * SLEN * sizeof(float)       // 256 KB dist^2 panel
                 + (size_t)SLEN * sizeof(unsigned long long);       //  32 KB sort keys

    GetStoneDistAngle3d_kernel<<<nblocks, TPB, shmem, stream>>>(coords, stones, out, N, S);
}